// Kalman_EKF_Gradient_54339926229233
// MI455X (gfx1250) — compile-verified
//
#include <hip/hip_runtime.h>

// ---------------------------------------------------------------------------
// Batched EKF loss on MI455X (gfx1250, wave32).
// The 6-state filter exactly decouples into three 2x2 scalar-measurement
// filters; one thread per (segment, subsystem), sub-major for wave-uniform
// control flow. Final reduction: per-wave shfl tree -> ws partials -> single
// wave that sums and collapses across lanes with v_wmma_f32_16x16x4_f32.
// ---------------------------------------------------------------------------

#define EKF_DT      (1.0f / 120.0f)
#define EKF_G       9.81f
#define EKF_KSIGN   100.0f
#define PI_1_5      4.71238898038469f
#define TWO_PI_F    6.2831853071795864f

typedef __attribute__((ext_vector_type(2))) float v2f;
typedef __attribute__((ext_vector_type(8))) float v8f;

__global__ __launch_bounds__(256) void ekf_sub_kernel(
    const float* __restrict__ params,
    const float* __restrict__ cp,
    const float* __restrict__ init_state,
    const float* __restrict__ meas,
    float* __restrict__ wave_partials,
    int N, int T)
{
    const int gid   = blockIdx.x * blockDim.x + threadIdx.x;
    const int total = 3 * N;
    float loss = 0.0f;

    if (gid < total) {
        const float fric = fabsf(params[0]);
        const float damp = fabsf(params[1]);
        const int sub = gid / N;          // 0 = X, 1 = Y, 2 = angle (uniform per block)
        const int n   = gid - sub * N;

        // Measurement component `sub` of segment n, one float per timestep.
        const float* zp = meas + (size_t)n * (size_t)T * 3 + sub;

        const float r = expf(cp[sub]);    // measurement noise for this subsystem
        float x0, x1;                     // (position-like, velocity-like)
        float p00 = 0.01f, p01 = 0.0f, p11 = 0.01f;   // symmetric 2x2 covariance

        if (sub < 2) {
            // Kinematic filter: friction via tanh, state-dependent Jacobian.
            const float q0 = expf(cp[3]);           // process noise (pos)
            const float q1 = expf(cp[4]);           // process noise (vel)
            const float fg = fric * EKF_G;
            x0 = init_state[n * 6 + sub];
            x1 = init_state[n * 6 + 2 + sub];
            for (int t = 0; t < T; ++t) {
                if (t + 24 < T) __builtin_prefetch(zp + (t + 24) * 3, 0, 0);
                const float z   = zp[t * 3];
                const float th  = tanhf(EKF_KSIGN * x1);
                const float x0n = fmaf(EKF_DT, x1, x0);
                const float x1n = x1 - EKF_DT * (damp * x1 + fg * th);
                const float dv  = 1.0f - EKF_DT * (damp + fg * EKF_KSIGN * (1.0f - th * th));
                // P_pred = F P F^T + Q  (F = [[1,DT],[0,dv]])
                const float pp00 = p00 + EKF_DT * (2.0f * p01 + EKF_DT * p11) + q0;
                const float pp01 = dv * (p01 + EKF_DT * p11);
                const float pp11 = dv * dv * p11 + q1;
                const float y  = z - x0n;
                const float S  = pp00 + r;
                const float Si = 1.0f / S;
                const float K0 = pp00 * Si;
                const float K1 = pp01 * Si;
                x0 = fmaf(K0, y, x0n);
                x1 = fmaf(K1, y, x1n);
                const float omK0 = 1.0f - K0;
                p00 = omK0 * pp00;
                p01 = omK0 * pp01;
                p11 = pp11 - K1 * pp01;
                loss += 0.5f * (logf(S) + y * y * Si);
            }
        } else {
            // Angle filter: constant Jacobian, +-2pi innovation wrap.
            const float q0  = expf(cp[5]);          // process noise (theta)
            const float q1  = expf(cp[6]);          // process noise (omega)
            const float dv  = 1.0f - EKF_DT * damp;
            const float dv2 = dv * dv;
            x0 = init_state[n * 6 + 4];
            x1 = init_state[n * 6 + 5];
            for (int t = 0; t < T; ++t) {
                if (t + 24 < T) __builtin_prefetch(zp + (t + 24) * 3, 0, 0);
                const float z   = zp[t * 3];
                const float x0n = fmaf(EKF_DT, x1, x0);
                const float x1n = x1 * dv;
                const float pp00 = p00 + EKF_DT * (2.0f * p01 + EKF_DT * p11) + q0;
                const float pp01 = dv * (p01 + EKF_DT * p11);
                const float pp11 = dv2 * p11 + q1;
                float y = z - x0n;
                y = (y > PI_1_5) ? (y - TWO_PI_F)
                                 : ((y < -PI_1_5) ? (y + TWO_PI_F) : y);
                const float S  = pp00 + r;
                const float Si = 1.0f / S;
                const float K0 = pp00 * Si;
                const float K1 = pp01 * Si;
                x0 = fmaf(K0, y, x0n);
                x1 = fmaf(K1, y, x1n);
                const float omK0 = 1.0f - K0;
                p00 = omK0 * pp00;
                p01 = omK0 * pp01;
                p11 = pp11 - K1 * pp01;
                loss += 0.5f * (logf(S) + y * y * Si);
            }
        }
    }

    // wave32 reduction -> one partial per wave (deterministic order)
    for (int off = 16; off > 0; off >>= 1)
        loss += __shfl_down(loss, off, 32);
    if ((threadIdx.x & 31) == 0)
        wave_partials[gid >> 5] = loss;
}

// Single-wave final reduction using the matrix unit.
// Per-lane value v goes into A(16x4): lanes 0-15 hold A[m][0] (VGPR0),
// lanes 16-31 hold A[m][2]; VGPR1 (K=1,3) is zero. B(4x16) = ones.
// D[i][j] = v[i] + v[i+16] for every column j; column 0 lives in VGPRs 0..7
// of lane 0 (M=0..7) and lane 16 (M=8..15), so 8 adds + one shfl_xor(16)
// complete the 32-lane sum exactly in f32.
__global__ __launch_bounds__(32) void reduce_wmma_kernel(
    const float* __restrict__ partials, int count, float scale,
    float* __restrict__ out)
{
    const int lane = threadIdx.x;
    float v = 0.0f;
    for (int i = lane; i < count; i += 32)
        v += partials[i];

    v2f a; a[0] = v;    a[1] = 0.0f;
    v2f b; b[0] = 1.0f; b[1] = 1.0f;
    v8f c = {};
    c = __builtin_amdgcn_wmma_f32_16x16x4_f32(
            /*neg_a=*/false, a, /*neg_b=*/false, b,
            /*c_mod=*/(short)0, c, /*reuse_a=*/false, /*reuse_b=*/false);

    float colsum = c[0] + c[1] + c[2] + c[3] + c[4] + c[5] + c[6] + c[7];
    colsum += __shfl_xor(colsum, 16, 32);
    if (lane == 0)
        out[0] = colsum * scale;
}

extern "C" void kernel_launch(void* const* d_in, const int* in_sizes, int n_in,
                              void* d_out, int out_size, void* d_ws, size_t ws_size,
                              hipStream_t stream) {
    const float* params     = (const float*)d_in[0];   // (4,)
    const float* cp         = (const float*)d_in[1];   // (7,)
    const float* init_state = (const float*)d_in[2];   // (N, 6)
    const float* meas       = (const float*)d_in[3];   // (N, T, 3)

    const int N = in_sizes[2] / 6;
    const int T = in_sizes[3] / (3 * N);

    const int total  = 3 * N;                  // one thread per (segment, subsystem)
    const int block  = 256;
    const int grid   = (total + block - 1) / block;
    const int nwaves = grid * (block / 32);

    float* partials = (float*)d_ws;            // nwaves floats (3 KB for N=8192)

    ekf_sub_kernel<<<grid, block, 0, stream>>>(params, cp, init_state, meas,
                                               partials, N, T);

    const float scale = 1.0f / (float)N;
    reduce_wmma_kernel<<<1, 32, 0, stream>>>(partials, nwaves, scale,
                                             (float*)d_out);
}